// Controller_19258633355377
// MI455X (gfx1250) — compile-verified
//
#include <hip/hip_runtime.h>

typedef __attribute__((ext_vector_type(2))) float v2f;
typedef __attribute__((ext_vector_type(8))) float v8f;
typedef __attribute__((ext_vector_type(4))) float f4;

#define N_NET 16
#define HID 32

// Single-wave kernel: 16 tiny MLPs evaluated for batch row 0 only.
// Layer 2 and layer 3 run on v_wmma_f32_16x16x4_f32 (full f32 precision).
__global__ __launch_bounds__(32) void fuzzy_head_kernel(
    const float* __restrict__ s,  const float* __restrict__ W1,
    const float* __restrict__ b1, const float* __restrict__ W2,
    const float* __restrict__ b2, const float* __restrict__ W3,
    const float* __restrict__ b3, float* __restrict__ ws)
{
    __shared__ float lds_h1[N_NET * HID];
    __shared__ float lds_h2[N_NET * HID];
    __shared__ float lds_dot[N_NET];
    __shared__ float lds_m[N_NET];

    const int lane = threadIdx.x;   // 0..31, one wave
    const int h    = lane >> 4;     // lane half (K offset selector)
    const int n    = lane & 15;     // N / M index within half

    // ---- Layer 1 (VALU): h1[k][o] = relu(s0[k%4]*W1[k][o] + b1[k][o]) ----
    const float sf[4] = { s[0], s[1], s[2], s[3] };
    #pragma unroll
    for (int k = 0; k < N_NET; ++k) {
        float v = sf[k & 3] * W1[k * HID + lane] + b1[k * HID + lane];
        lds_h1[k * HID + lane] = fmaxf(v, 0.0f);
    }
    __syncthreads();

    // ---- Layer 2 (WMMA): per network, h2(1x32) = h1(1x32) @ W2(32x32) ----
    // A: all 16 M-rows replicated with the h1 K-chunk (C rows all identical).
    // A/B K mapping: K = vgpr + 2*laneHalf  (B layout = A layout of B^T).
    for (int k = 0; k < N_NET; ++k) {
        for (int nt = 0; nt < 2; ++nt) {
            v8f acc = {};
            const float* w2 = W2 + k * HID * HID + nt * 16 + n;
            #pragma unroll
            for (int kc = 0; kc < 8; ++kc) {
                const int kb = 4 * kc + 2 * h;
                v2f a, b;
                a.x = lds_h1[k * HID + kb];
                a.y = lds_h1[k * HID + kb + 1];
                b.x = w2[kb * HID];          // W2[k][kb  ][nt*16+n]
                b.y = w2[(kb + 1) * HID];    // W2[k][kb+1][nt*16+n]
                acc = __builtin_amdgcn_wmma_f32_16x16x4_f32(
                        false, a, false, b, (short)0, acc, false, false);
            }
            // C row M=0 lives in acc[0]; every lane holds column N = n.
            float h2v = fmaxf(acc[0] + b2[k * HID + nt * 16 + n], 0.0f);
            if (h == 0) lds_h2[k * HID + nt * 16 + n] = h2v;
        }
    }
    __syncthreads();

    // ---- Layer 3 (WMMA): C = H2(16x32) * W3^T -> diagonal = per-net dots ----
    {
        v8f acc = {};
        #pragma unroll
        for (int kc = 0; kc < 8; ++kc) {
            const int kb = 4 * kc + 2 * h;
            v2f a, b;
            a.x = lds_h2[n * HID + kb];      // A[m=n][kb]
            a.y = lds_h2[n * HID + kb + 1];
            b.x = W3[n * HID + kb];          // B[kb][n'] = W3[net n'][kb]
            b.y = W3[n * HID + kb + 1];
            acc = __builtin_amdgcn_wmma_f32_16x16x4_f32(
                    false, a, false, b, (short)0, acc, false, false);
        }
        // diagonal k = M = N: VGPR j with j + 8*h == n holds C[k,k], k = n.
        #pragma unroll
        for (int j = 0; j < 8; ++j)
            if (j + 8 * h == n) lds_dot[n] = acc[j];
    }
    __syncthreads();

    // ---- sigmoid, means over 8, softmax over 2 actions ----
    if (lane < N_NET) {
        float z = lds_dot[lane] + b3[lane];
        lds_m[lane] = 1.0f / (1.0f + expf(-z));
    }
    __syncthreads();

    if (lane == 0) {
        float sA = 0.0f, sB = 0.0f;
        #pragma unroll
        for (int i = 0; i < 8; ++i) { sA += lds_m[i]; sB += lds_m[8 + i]; }
        sA *= 0.125f; sB *= 0.125f;                 // mean over (rules, members)
        float mx  = fmaxf(sA, sB);
        float eA  = expf(sA - mx), eB = expf(sB - mx);
        float inv = 1.0f / (eA + eB);
        ws[0] = eA * inv;
        ws[1] = eB * inv;
    }
}

// Broadcast (p0,p1) to all B rows: pure b128 stores, HBM-write bound (~2MB).
__global__ void broadcast_fill_kernel(const float* __restrict__ ws,
                                      f4* __restrict__ out, int nvec)
{
    int i = blockIdx.x * blockDim.x + threadIdx.x;
    float p0 = ws[0], p1 = ws[1];
    f4 v = { p0, p1, p0, p1 };
    if (i < nvec) out[i] = v;
}

extern "C" void kernel_launch(void* const* d_in, const int* in_sizes, int n_in,
                              void* d_out, int out_size, void* d_ws, size_t ws_size,
                              hipStream_t stream) {
    const float* s  = (const float*)d_in[0];
    const float* W1 = (const float*)d_in[1];
    const float* b1 = (const float*)d_in[2];
    const float* W2 = (const float*)d_in[3];
    const float* b2 = (const float*)d_in[4];
    const float* W3 = (const float*)d_in[5];
    const float* b3 = (const float*)d_in[6];
    float* ws  = (float*)d_ws;
    float* out = (float*)d_out;

    fuzzy_head_kernel<<<1, 32, 0, stream>>>(s, W1, b1, W2, b2, W3, b3, ws);

    int nvec    = out_size / 4;                 // float4 per 2 batch rows
    int threads = 256;
    int blocks  = (nvec + threads - 1) / threads;
    broadcast_fill_kernel<<<blocks, threads, 0, stream>>>(ws, (f4*)out, nvec);
}